// YoloX_78958678770193
// MI455X (gfx1250) — compile-verified
//
#include <hip/hip_runtime.h>
#include <stdint.h>

typedef __attribute__((ext_vector_type(4))) unsigned int v4u;
typedef __attribute__((ext_vector_type(8))) int v8i;
typedef __attribute__((ext_vector_type(4))) int v4i;

#define BATCH   64
#define CHANS   85
#define NPTS    8400
#define TILE_N  256
#define LDS_STRIDE 257   // 256 dwords + 1 pad dword (TDM pad) -> 257 % 64 == 1, conflict-free column reads
#define N_TILES ((NPTS + TILE_N - 1) / TILE_N)  // 33

__global__ __launch_bounds__(256) void yolox_decode_kernel(
    const float* __restrict__ dt, const float* __restrict__ anchors,
    float* __restrict__ out)
{
  __shared__ float tile[CHANS * LDS_STRIDE];

  const int tid      = threadIdx.x;
  const int tile_idx = blockIdx.x;
  const int b        = blockIdx.y;
  const int n0       = tile_idx * TILE_N;
  const int rem      = min(TILE_N, NPTS - n0);

  // ---------------- TDM: async 2D tile load (rem x 85, zero-padded to 256 x 85) into LDS ----------------
  if (tid < 32) {  // wave 0 issues the tensor DMA (TDM ignores EXEC; one issue per workgroup)
    uint64_t gaddr    = (uint64_t)(const void*)(dt + (size_t)b * CHANS * NPTS + n0);
    uint32_t lds_base = (uint32_t)(uint64_t)(&tile[0]);

    v4u g0;
    g0[0] = 1u;                                      // count=1 (valid), user mode, no gather
    g0[1] = lds_base;                                // lds_addr (bytes)
    g0[2] = (uint32_t)(gaddr & 0xFFFFFFFFu);         // global_addr[31:0] (tile start)
    g0[3] = (uint32_t)((gaddr >> 32) & 0x1FFFFFFu)   // global_addr[56:32]
          | (2u << 30);                              // type = 2 ("image")

    v8i g1;
    g1[0] = (int)((2u << 16)                         // data_size = 4 bytes
                | (1u << 20)                         // pad_enable
                | (7u << 22));                       // pad_interval = 256 dwords (exactly one row)
                                                     // pad_amount = 0 -> 1 dword pad per row
    g1[1] = (int)(((uint32_t)rem & 0xFFFFu) << 16);  // tensor_dim0 lo16 (x OOB limit -> tail reads return 0)
    g1[2] = (int)((((uint32_t)rem >> 16) & 0xFFFFu)  // tensor_dim0 hi16
                | ((uint32_t)CHANS << 16));          // tensor_dim1 lo16 = 85
    g1[3] = (int)((uint32_t)TILE_N << 16);           // tensor_dim1 hi16 = 0 | tile_dim0 = 256
    g1[4] = (int)CHANS;                              // tile_dim1 = 85, tile_dim2 = 0 (2D)
    g1[5] = NPTS;                                    // tensor_dim0_stride lo32 = 8400 (row-to-row in dt)
    g1[6] = 0;                                       // stride hi16 / dim1_stride lo16
    g1[7] = 0;                                       // dim1_stride hi32 (unused, 2D)

    v4i g2 = {0, 0, 0, 0};                           // dims 2/3 unused
    v4i g3 = {0, 0, 0, 0};                           // dim 4 unused
    v8i g4 = {0, 0, 0, 0, 0, 0, 0, 0};               // extra arg of the 6-arg builtin (zero-filled)

    __builtin_amdgcn_tensor_load_to_lds(g0, g1, g2, g3, g4, 0);
    __builtin_amdgcn_s_wait_tensorcnt(0);
  }
  __syncthreads();

  // ---------------- transform + transposed, vectorized coalesced store ----------------
  const int valid = rem * CHANS;                     // output elements in this tile (contiguous!)
  float* out_base = out + ((size_t)b * NPTS + n0) * CHANS;

  for (int i4 = tid * 4; i4 < valid; i4 += 256 * 4) {
    float vals[4];
#pragma unroll
    for (int j = 0; j < 4; ++j) {
      const int i = i4 + j;
      if (i < valid) {
        const int n_local = i / CHANS;
        const int c       = i - n_local * CHANS;
        const float v     = tile[c * LDS_STRIDE + n_local];
        const int n       = n0 + n_local;
        float r;
        if (c < 2) {
          // xy = anchor_xy + v * anchor_wh
          r = anchors[n * 4 + c] + v * anchors[n * 4 + c + 2];
        } else if (c < 4) {
          // wh = anchor_wh * exp(v)
          r = anchors[n * 4 + c] * __expf(v);
        } else {
          // scores = sigmoid(v)
          r = 1.0f / (1.0f + __expf(-v));
        }
        vals[j] = r;
      } else {
        vals[j] = 0.0f;
      }
    }
    if (i4 + 3 < valid) {
      // out_base + i4 is 16B aligned: (b*NPTS + n0)*85*4 and i4*4 are multiples of 16
      *(float4*)(out_base + i4) = make_float4(vals[0], vals[1], vals[2], vals[3]);
    } else {
      for (int j = 0; j < 4 && i4 + j < valid; ++j) out_base[i4 + j] = vals[j];
    }
  }
}

extern "C" void kernel_launch(void* const* d_in, const int* in_sizes, int n_in,
                              void* d_out, int out_size, void* d_ws, size_t ws_size,
                              hipStream_t stream) {
  (void)in_sizes; (void)n_in; (void)out_size; (void)d_ws; (void)ws_size;
  const float* dt      = (const float*)d_in[0];
  const float* anchors = (const float*)d_in[1];
  float* out           = (float*)d_out;

  dim3 grid(N_TILES, BATCH, 1);
  yolox_decode_kernel<<<grid, 256, 0, stream>>>(dt, anchors, out);
}